// StreamingAttentionSink_69080253988897
// MI455X (gfx1250) — compile-verified
//
#include <hip/hip_runtime.h>
#include <math.h>
#include <stdint.h>

typedef float v2f __attribute__((ext_vector_type(2)));
typedef float v4f __attribute__((ext_vector_type(4)));
typedef float v8f __attribute__((ext_vector_type(8)));

#define NHEADS   32
#define NKVH     8
#define GQA      4
#define HDIM     128
#define BLKSZ    16
#define MAXBLK   256
#define SEQCAP   4096   /* T */
#define NBATCH   16
#define MODELLEN 4096   /* L */
#define NSPLIT   16
#define TILES_PER_SPLIT (MAXBLK / NSPLIT)  /* 16 */
#define LDST     132    /* LDS row stride in floats: rows 16B aligned */
#define SSTRIDE  20
#define ATTN_SCALE 0.08838834764831845f    /* 1/sqrt(128) */

// ---------------------------------------------------------------------------
// Pass 0: RoPE cos/sin tables for positions 0..T-1 (2 MB, L2 resident)
// ---------------------------------------------------------------------------
__global__ __launch_bounds__(64)
void rope_table_kernel(float* __restrict__ cosT, float* __restrict__ sinT) {
  const int t = blockIdx.x;     // 0..4095
  const int i = threadIdx.x;    // 0..63
  float inv = powf(10000.0f, -((float)i) / 64.0f);
  float ang = (float)t * inv;
  cosT[(size_t)t * 64 + i] = cosf(ang);
  sinT[(size_t)t * 64 + i] = sinf(ang);
}

// LDS byte offset of a __shared__ pointer: flat LDS aperture keeps the
// wave-relative offset in addr[31:0] (ISA 10.2 aperture mapping).
__device__ __forceinline__ uint32_t lds_off(const void* p) {
  return (uint32_t)(uintptr_t)p;
}

// Issue one 16-token tile (K + V) as 32 async b128 global->LDS copies.
// Each instruction moves one 512B row (32 lanes x 16B).
__device__ __forceinline__ void prefetch_tile_async(
    const float* __restrict__ kcache, const float* __restrict__ vcache,
    int phys, int kvh, uint32_t kldsBase, uint32_t vldsBase, int lane)
{
  const size_t rowStride = (size_t)NKVH * HDIM;       // floats
  const float* kr = kcache + ((size_t)phys * BLKSZ) * rowStride
                           + (size_t)kvh * HDIM + 4 * lane;
  const float* vr = vcache + ((size_t)phys * BLKSZ) * rowStride
                           + (size_t)kvh * HDIM + 4 * lane;
  // async writes must not race in-flight ds ops on the target buffer
  asm volatile("s_wait_dscnt 0x0" ::: "memory");
  #pragma unroll
  for (int j = 0; j < 16; ++j) {
    uint64_t kg = (uint64_t)(uintptr_t)(kr + (size_t)j * rowStride);
    uint64_t vg = (uint64_t)(uintptr_t)(vr + (size_t)j * rowStride);
    uint32_t kl = kldsBase + (uint32_t)(j * LDST + 4 * lane) * 4u;
    uint32_t vl = vldsBase + (uint32_t)(j * LDST + 4 * lane) * 4u;
    asm volatile("global_load_async_to_lds_b128 %0, %1, off"
                 :: "v"(kl), "v"(kg) : "memory");
    asm volatile("global_load_async_to_lds_b128 %0, %1, off"
                 :: "v"(vl), "v"(vg) : "memory");
  }
}

// ---------------------------------------------------------------------------
// Pass 1: flash-decode partials. One wave32 per (split, kvh, batch).
// Double-buffered async global->LDS pipeline.
// ---------------------------------------------------------------------------
__global__ __launch_bounds__(32)
void attn_kernel(const float* __restrict__ q,      // [B, H*D]
                 const float* __restrict__ knew,   // [B, KVH*D]
                 const float* __restrict__ vnew,   // [B, KVH*D]
                 const float* __restrict__ kcache, // [NB, BS, KVH, D]
                 const float* __restrict__ vcache, // [NB, BS, KVH, D]
                 const int*   __restrict__ btab,   // [B, MAXBLK]
                 const int*   __restrict__ ctxlen, // [B]
                 const float* __restrict__ cosT,
                 const float* __restrict__ sinT,
                 float* __restrict__ partO,        // [B,KVH,G,NSPLIT,D]
                 float* __restrict__ partML)       // [B,KVH,G,NSPLIT,2]
{
  __shared__ float kB[2][16][LDST];  // K tiles (raw -> roped in place)
  __shared__ float vB[2][16][LDST];  // V tiles (used raw)
  __shared__ float qs[16][LDST];     // scaled+roped Q, rows>=4 zero
  __shared__ float sT[16][SSTRIDE];  // P tile [head][token], rows>=4 zero
  __shared__ float knR[HDIM];
  __shared__ float vnR[HDIM];
  __shared__ float hscale[8];

  const int lane  = threadIdx.x;
  const int l16   = lane & 15;
  const int hi    = lane >> 4;
  const int split = blockIdx.x;
  const int kvh   = blockIdx.y;
  const int b     = blockIdx.z;

  const int ctx     = ctxlen[b];
  const int rem     = ctx % BLKSZ;
  const int seq_len = (ctx >= MODELLEN) ? (MODELLEN - BLKSZ + rem + 1) : (ctx + 1);
  const int pos_q   = seq_len - 1;
  const int numTiles = (seq_len + BLKSZ - 1) / BLKSZ;
  const int tBeg = split * TILES_PER_SPLIT;
  int tEnd = tBeg + TILES_PER_SPLIT;
  if (tEnd > numTiles) tEnd = numTiles;
  const int nt = tEnd - tBeg;

  const uint32_t kOff[2] = { lds_off(&kB[0][0][0]), lds_off(&kB[1][0][0]) };
  const uint32_t vOff[2] = { lds_off(&vB[0][0][0]), lds_off(&vB[1][0][0]) };

  // kick off the first tile's async copy before doing any preamble work
  if (nt > 0) {
    prefetch_tile_async(kcache, vcache, btab[b * MAXBLK + tBeg], kvh,
                        kOff[0], vOff[0], lane);
  }

  // ---- preamble -----------------------------------------------------------
  for (int idx = lane; idx < 16 * LDST;    idx += 32) ((float*)qs)[idx] = 0.0f;
  for (int idx = lane; idx < 16 * SSTRIDE; idx += 32) ((float*)sT)[idx] = 0.0f;

  // roped + scaled Q for this block's 4 heads (256 rope pairs)
  for (int p = 0; p < 8; ++p) {
    int pr = lane * 8 + p;
    int g  = pr >> 6;
    int dd = pr & 63;
    float c = cosT[(size_t)pos_q * 64 + dd];
    float s = sinT[(size_t)pos_q * 64 + dd];
    const float* qp = q + ((size_t)b * NHEADS + kvh * GQA + g) * HDIM;
    float q1 = qp[dd], q2 = qp[dd + 64];
    qs[g][dd]      = (q1 * c - q2 * s) * ATTN_SCALE;
    qs[g][dd + 64] = (q2 * c + q1 * s) * ATTN_SCALE;
  }
  // roped current-token key (64 pairs)
  for (int p = 0; p < 2; ++p) {
    int dd = lane * 2 + p;
    float c = cosT[(size_t)pos_q * 64 + dd];
    float s = sinT[(size_t)pos_q * 64 + dd];
    const float* kp = knew + ((size_t)b * NKVH + kvh) * HDIM;
    float k1 = kp[dd], k2 = kp[dd + 64];
    knR[dd]      = k1 * c - k2 * s;
    knR[dd + 64] = k2 * c + k1 * s;
  }
  {
    const float* vp = vnew + ((size_t)b * NKVH + kvh) * HDIM;
    for (int i = 0; i < 4; ++i) vnR[lane * 4 + i] = vp[lane * 4 + i];
  }
  __syncthreads();

  float mRun = -3.0e38f, lRun = 0.0f;  // lanes 0..3 carry head state
  v8f acc[8];
  for (int c = 0; c < 8; ++c)
    for (int r = 0; r < 8; ++r) acc[c][r] = 0.0f;

  // ---- main token-tile loop (double-buffered) -----------------------------
  for (int ii = 0; ii < nt; ++ii) {
    const int tile = tBeg + ii;
    const int t0   = tile * BLKSZ;
    const int cur  = ii & 1;

    // prefetch next tile into the other buffer
    if (ii + 1 < nt) {
      prefetch_tile_async(kcache, vcache, btab[b * MAXBLK + tile + 1], kvh,
                          kOff[cur ^ 1], vOff[cur ^ 1], lane);
      asm volatile("s_wait_asynccnt 0x20" ::: "memory");  // tile `ii` landed
    } else {
      asm volatile("s_wait_asynccnt 0x0" ::: "memory");
    }

    float (*kT)[LDST] = kB[cur];
    float (*vT)[LDST] = vB[cur];

    // RoPE K tile in place: 2 lanes per token row (one per 32-pair half)
    {
      const int row  = lane >> 1;
      const int half = lane & 1;
      const size_t tb = (size_t)(t0 + row) * 64;
      const int d0 = half * 32;
      for (int p = 0; p < 32; p += 4) {
        v4f k1 = *(const v4f*)&kT[row][d0 + p];
        v4f k2 = *(const v4f*)&kT[row][d0 + p + 64];
        v4f cc = *(const v4f*)(cosT + tb + d0 + p);
        v4f ss = *(const v4f*)(sinT + tb + d0 + p);
        *(v4f*)&kT[row][d0 + p]      = k1 * cc - k2 * ss;
        *(v4f*)&kT[row][d0 + p + 64] = k2 * cc + k1 * ss;
      }
    }
    // substitute current token's k (already roped at pos_q) and v
    if (pos_q >= t0 && pos_q < t0 + BLKSZ) {
      __syncthreads();
      const int row = pos_q - t0;
      for (int i = 0; i < 4; ++i) {
        kT[row][lane * 4 + i] = knR[lane * 4 + i];
        vT[row][lane * 4 + i] = vnR[lane * 4 + i];
      }
    }
    __syncthreads();

    // QK^T: D[token][head] = sum_d K[token][d] * Qs[head][d]
    // A (16x4 f32): lane holds A[l16][4kk+2hi+{0,1}]; B (4x16): B[2hi+j][l16]
    v8f sc;
    for (int r = 0; r < 8; ++r) sc[r] = 0.0f;
    for (int kk = 0; kk < 32; ++kk) {
      v2f a  = *(const v2f*)&kT[l16][4 * kk + 2 * hi];
      v2f bq = *(const v2f*)&qs[l16][4 * kk + 2 * hi];
      sc = __builtin_amdgcn_wmma_f32_16x16x4_f32(false, a, false, bq,
                                                 (short)0, sc, false, false);
    }

    // mask + stage scores to LDS as sT[head][token] (heads 0..3 only)
    for (int r = 0; r < 8; ++r) {
      int m = r + 8 * hi;                 // token within tile
      float sv = sc[r];
      if (t0 + m >= seq_len) sv = -3.0e38f;
      if (l16 < GQA) sT[l16][m] = sv;
    }
    __syncthreads();

    // online softmax per head (lanes 0..3 serially over 16 tokens)
    if (lane < GQA) {
      float tmax = -3.0e38f;
      for (int t = 0; t < 16; ++t) tmax = fmaxf(tmax, sT[lane][t]);
      float mNew = fmaxf(mRun, tmax);
      float scl  = expf(mRun - mNew);
      float lNew = lRun * scl;
      for (int t = 0; t < 16; ++t) {
        float p = expf(sT[lane][t] - mNew);
        sT[lane][t] = p;
        lNew += p;
      }
      mRun = mNew; lRun = lNew;
      hscale[lane] = scl;
    }
    if (lane >= GQA && lane < 8) hscale[lane] = 1.0f;
    __syncthreads();

    // rescale accumulators by per-head exp(mOld - mNew)
    {
      float hs[8];
      for (int r = 0; r < 8; ++r) hs[r] = hscale[r];
      for (int c = 0; c < 8; ++c)
        for (int r = 0; r < 8; ++r) acc[c][r] *= hs[r];
    }

    // PV: D[head][dim] += sum_t P[head][t] * V[t][dim]  (8 dim-chunks of 16)
    for (int c = 0; c < 8; ++c) {
      for (int kk = 0; kk < 4; ++kk) {
        v2f ap = *(const v2f*)&sT[l16][4 * kk + 2 * hi];
        v2f bv;
        bv[0] = vT[4 * kk + 2 * hi + 0][c * 16 + l16];
        bv[1] = vT[4 * kk + 2 * hi + 1][c * 16 + l16];
        acc[c] = __builtin_amdgcn_wmma_f32_16x16x4_f32(false, ap, false, bv,
                                                       (short)0, acc[c], false, false);
      }
    }
  }

  // ---- write partials -----------------------------------------------------
  if (lane < GQA) {
    size_t idx = (((size_t)b * NKVH + kvh) * GQA + lane) * NSPLIT + split;
    partML[idx * 2]     = mRun;
    partML[idx * 2 + 1] = lRun;
  }
  if (hi == 0) { // lanes 0..15 hold heads r=0..7 (only 0..3 real)
    for (int r = 0; r < GQA; ++r) {
      size_t base = ((((size_t)b * NKVH + kvh) * GQA + r) * NSPLIT + split) * (size_t)HDIM;
      for (int c = 0; c < 8; ++c)
        partO[base + c * 16 + l16] = acc[c][r];
    }
  }
}

// ---------------------------------------------------------------------------
// Pass 2: merge split partials (log-sum-exp combine)
// ---------------------------------------------------------------------------
__global__ __launch_bounds__(128)
void reduce_kernel(const float* __restrict__ partO,
                   const float* __restrict__ partML,
                   float* __restrict__ out)
{
  const int d   = threadIdx.x;   // 0..127
  const int h   = blockIdx.x;    // 0..31
  const int b   = blockIdx.y;    // 0..15
  const int kvh = h >> 2;
  const int g   = h & 3;
  size_t base = (((size_t)b * NKVH + kvh) * GQA + g) * NSPLIT;
  float M = -3.0e38f;
  for (int i = 0; i < NSPLIT; ++i) M = fmaxf(M, partML[(base + i) * 2]);
  float L = 0.0f, o = 0.0f;
  for (int i = 0; i < NSPLIT; ++i) {
    float w = expf(partML[(base + i) * 2] - M);
    L += w * partML[(base + i) * 2 + 1];
    o += w * partO[(base + i) * (size_t)HDIM + d];
  }
  out[((size_t)b * NHEADS + h) * HDIM + d] = o / fmaxf(L, 1e-30f);
}

// ---------------------------------------------------------------------------
extern "C" void kernel_launch(void* const* d_in, const int* in_sizes, int n_in,
                              void* d_out, int out_size, void* d_ws, size_t ws_size,
                              hipStream_t stream) {
  const float* q      = (const float*)d_in[0];
  const float* k      = (const float*)d_in[1];
  const float* v      = (const float*)d_in[2];
  const float* kcache = (const float*)d_in[3];
  const float* vcache = (const float*)d_in[4];
  const int*   btab   = (const int*)d_in[5];
  const int*   ctxlen = (const int*)d_in[6];
  float* outp = (float*)d_out;

  float* ws     = (float*)d_ws;
  float* cosT   = ws;                                   // 4096*64
  float* sinT   = cosT + (size_t)SEQCAP * 64;           // 4096*64
  float* partO  = sinT + (size_t)SEQCAP * 64;           // B*KVH*G*NSPLIT*128
  float* partML = partO + (size_t)NBATCH * NKVH * GQA * NSPLIT * HDIM; // *2

  rope_table_kernel<<<dim3(SEQCAP), dim3(64), 0, stream>>>(cosT, sinT);
  attn_kernel<<<dim3(NSPLIT, NKVH, NBATCH), dim3(32), 0, stream>>>(
      q, k, v, kcache, vcache, btab, ctxlen, cosT, sinT, partO, partML);
  reduce_kernel<<<dim3(NHEADS, NBATCH), dim3(128), 0, stream>>>(partO, partML, outp);
}